// DualAssignmentLoss_17162689314991
// MI455X (gfx1250) — compile-verified
//
#include <hip/hip_runtime.h>

#define EPS_IOU_F 1e-7f
#define EPS_AS_F  1e-9f
#define INV_PI2_4 0.4052847345693511f  // 4 / pi^2

#define Bb 16
#define Nn 8400
#define Mm 64
#define THREADS 256
#define NWAVES (THREADS / 32)
#define CHUNKS ((Nn + THREADS - 1) / THREADS)   // 33
#define NBLK (Bb * CHUNKS)                      // 528

typedef __attribute__((ext_vector_type(2))) float v2f;
typedef __attribute__((ext_vector_type(8))) float v8f;

// ds_swizzle group-of-32 xor pattern: and=0x1F, or=0, xor=mask
#define SWZ_IMM(mask) (0x1F | ((mask) << 10))
#define SWZ_ADDF(s, mask)                                                     \
  (s) += __int_as_float(__builtin_amdgcn_ds_swizzle(__float_as_int(s), SWZ_IMM(mask)));

// ---------------------------------------------------------------------------
// 3-channel block sum through the matrix pipe.
// One V_WMMA_F32_16X16X4_F32 with a selector A-matrix reduces TWO channels
// 32->16 lanes at once:
//   A row0 = [1,0,1,0], row1 = [0,1,0,1]  (rows >=2 zero)
//   B lane j carries (c0,c1) in its two K slots, so
//   D(0,j) = c0[j]+c0[j+16],  D(1,j) = c1[j]+c1[j+16].
// d[0]/d[1] live in lanes 0..15 (lanes 16..31 hold zero rows 8/9); four
// xor ds_swizzle adds inside the 16-lane half finish each sum on lane 0.
// Results valid on thread 0 only.
// ---------------------------------------------------------------------------
__device__ __forceinline__ void block_sum3(float c0, float c1, float c2,
                                           float* sred,
                                           float& o0, float& o1, float& o2) {
  const int tid = threadIdx.x;
  const int lane = tid & 31, wid = tid >> 5, l16 = lane & 15;

  v2f a; a[0] = (l16 == 0) ? 1.f : 0.f; a[1] = (l16 == 1) ? 1.f : 0.f;
  v2f b01; b01[0] = c0; b01[1] = c1;
  v2f b2;  b2[0]  = c2; b2[1]  = 0.f;
  v8f z = {0.f,0.f,0.f,0.f,0.f,0.f,0.f,0.f};

  v8f d01 = __builtin_amdgcn_wmma_f32_16x16x4_f32(false, a, false, b01,
                                                  (short)0, z, false, false);
  v8f d2  = __builtin_amdgcn_wmma_f32_16x16x4_f32(false, a, false, b2,
                                                  (short)0, z, false, false);
  float s0 = d01[0], s1 = d01[1], s2 = d2[0];
  SWZ_ADDF(s0, 1) SWZ_ADDF(s0, 2) SWZ_ADDF(s0, 4) SWZ_ADDF(s0, 8)
  SWZ_ADDF(s1, 1) SWZ_ADDF(s1, 2) SWZ_ADDF(s1, 4) SWZ_ADDF(s1, 8)
  SWZ_ADDF(s2, 1) SWZ_ADDF(s2, 2) SWZ_ADDF(s2, 4) SWZ_ADDF(s2, 8)

  if (lane == 0) { sred[wid*3+0] = s0; sred[wid*3+1] = s1; sred[wid*3+2] = s2; }
  __syncthreads();
  o0 = o1 = o2 = 0.f;
  if (tid == 0) {
    #pragma unroll
    for (int i = 0; i < NWAVES; ++i) {
      o0 += sred[i*3+0]; o1 += sred[i*3+1]; o2 += sred[i*3+2];
    }
  }
  __syncthreads();   // protect sred for back-to-back calls
}

// ---------------------------------------------------------------------------
// math helpers (exact mirrors of the reference)
// ---------------------------------------------------------------------------
__device__ __forceinline__ float pair_iou(float px1, float py1, float px2, float py2,
                                          float gx1, float gy1, float gx2, float gy2) {
  float a1 = fmaxf((px2 - px1) * (py2 - py1), EPS_IOU_F);
  float a2 = fmaxf((gx2 - gx1) * (gy2 - gy1), EPS_IOU_F);
  float iw = fminf(px2, gx2) - fmaxf(px1, gx1);
  float ih = fminf(py2, gy2) - fmaxf(py1, gy1);
  float inter = fmaxf(iw, 0.f) * fmaxf(ih, 0.f);
  return inter / (a1 + a2 - inter + EPS_IOU_F);
}

__device__ __forceinline__ float ciou_loss_f(float4 p, float4 t) {
  float ix1 = fmaxf(p.x, t.x), iy1 = fmaxf(p.y, t.y);
  float ix2 = fminf(p.z, t.z), iy2 = fminf(p.w, t.w);
  float inter = fmaxf(ix2 - ix1, 0.f) * fmaxf(iy2 - iy1, 0.f);
  float pa = fmaxf((p.z - p.x) * (p.w - p.y), EPS_IOU_F);
  float ta = fmaxf((t.z - t.x) * (t.w - t.y), EPS_IOU_F);
  float iou = inter / (pa + ta - inter + EPS_IOU_F);
  float ex1 = fminf(p.x, t.x), ey1 = fminf(p.y, t.y);
  float ex2 = fmaxf(p.z, t.z), ey2 = fmaxf(p.w, t.w);
  float c2 = (ex2 - ex1) * (ex2 - ex1) + (ey2 - ey1) * (ey2 - ey1) + EPS_IOU_F;
  float dx = (p.x + p.z) * 0.5f - (t.x + t.z) * 0.5f;
  float dy = (p.y + p.w) * 0.5f - (t.y + t.w) * 0.5f;
  float rho2 = dx * dx + dy * dy;
  float pw = fmaxf(p.z - p.x, EPS_IOU_F), ph = fmaxf(p.w - p.y, EPS_IOU_F);
  float tw = fmaxf(t.z - t.x, EPS_IOU_F), th = fmaxf(t.w - t.y, EPS_IOU_F);
  float da = atanf(tw / th) - atanf(pw / ph);
  float v = INV_PI2_4 * da * da;
  float alpha = v / (1.f - iou + v + EPS_IOU_F);   // stop_gradient: fwd value only
  float l = 1.f - (iou - rho2 / c2 - alpha * v);
  return fminf(fmaxf(l, 0.f), 10.f);
}

__device__ __forceinline__ float vfl(float logit, float tgt) {
  float p = 1.f / (1.f + __expf(-logit));
  float bce = fmaxf(logit, 0.f) - logit * tgt + log1pf(__expf(-fabsf(logit)));
  float pos = (tgt > 0.f) ? 1.f : 0.f;
  float dd = fabsf(p - tgt);
  float w = tgt * pos + 0.75f * dd * dd * (1.f - pos);  // gamma==2 -> square
  return bce * w;
}

__device__ __forceinline__ float metric_of(float sc, float iou) {
  float ic = fminf(fmaxf(iou, 0.f), 1.f);
  float i2 = ic * ic;
  return sc * i2 * i2 * i2;                  // s^1 * iou^6
}

// ---------------------------------------------------------------------------
// Kernel A: per-(b,m) k-th largest alignment metric over N preds.
// Per-thread sorted top-K in registers (compare-exchange insertion), then K
// rounds of block-wide max extraction: 5-step ds_swizzle lexicographic
// (value,tid) wave max + one barrier per round (parity double-buffered LDS).
// ---------------------------------------------------------------------------
#define SWZ_MAX64(mask)                                                        \
  { unsigned ohi = (unsigned)__builtin_amdgcn_ds_swizzle((int)hi, SWZ_IMM(mask)); \
    unsigned olo = (unsigned)__builtin_amdgcn_ds_swizzle((int)lo, SWZ_IMM(mask)); \
    bool take = (ohi > hi) || ((ohi == hi) && (olo > lo));                     \
    hi = take ? ohi : hi; lo = take ? olo : lo; }

template <int K>
__global__ void topk_thr_kernel(const float* __restrict__ logits,
                                const float* __restrict__ boxes,
                                const float* __restrict__ gt,
                                const int*   __restrict__ mask,
                                float*       __restrict__ thr) {
  __shared__ unsigned long long wred[2][NWAVES];
  int blk = blockIdx.x;
  int b = blk >> 6, m = blk & 63;
  int gi = b * Mm + m;
  int tid = threadIdx.x;
  if (mask[gi] <= 0) { if (tid == 0) thr[gi] = 0.f; return; }

  float gx1 = gt[gi * 4 + 0], gy1 = gt[gi * 4 + 1];
  float gx2 = gt[gi * 4 + 2], gy2 = gt[gi * 4 + 3];

  float tk[K];
  #pragma unroll
  for (int j = 0; j < K; ++j) tk[j] = -1.f;

  const float4* boxes4 = (const float4*)boxes;
  for (int n = tid; n < Nn; n += THREADS) {
    int idx = b * Nn + n;
    if (n + THREADS < Nn) __builtin_prefetch(&boxes4[idx + THREADS], 0, 0);
    float lg = logits[idx];
    float4 pb = boxes4[idx];
    float sc = fminf(fmaxf(1.f / (1.f + __expf(-lg)), EPS_AS_F), 1.f);
    float v = metric_of(sc, pair_iou(pb.x, pb.y, pb.z, pb.w, gx1, gy1, gx2, gy2));
    // sorted-descending insertion, register-resident (static indices only)
    #pragma unroll
    for (int j = 0; j < K; ++j) {
      float mx = fmaxf(tk[j], v);
      float mn = fminf(tk[j], v);
      tk[j] = mx; v = mn;
    }
  }

  const int lane = tid & 31, wid = tid >> 5;
  float got = 0.f;
  for (int it = 0; it < K; ++it) {
    float lv = tk[0];                                      // local max (sorted)
    unsigned hi = (lv > 0.f) ? __float_as_uint(lv) : 0u;   // order-preserving
    unsigned lo = (unsigned)tid;
    SWZ_MAX64(1) SWZ_MAX64(2) SWZ_MAX64(4) SWZ_MAX64(8) SWZ_MAX64(16)
    if (lane == 0)
      wred[it & 1][wid] = (((unsigned long long)hi) << 32) | (unsigned long long)lo;
    __syncthreads();
    unsigned long long w = wred[it & 1][0];
    #pragma unroll
    for (int i = 1; i < NWAVES; ++i) {
      unsigned long long o = wred[it & 1][i];
      if (o > w) w = o;
    }
    if ((unsigned)(w & 0xFFFFFFFFull) == (unsigned)tid) {  // winner pops
      #pragma unroll
      for (int j = 0; j < K - 1; ++j) tk[j] = tk[j + 1];
      tk[K - 1] = -1.f;
    }
    got = __uint_as_float((unsigned)(w >> 32));
  }
  if (tid == 0) thr[gi] = got;                             // k-th largest (>=0)
}

// ---------------------------------------------------------------------------
// o2m head: fused assign + varifocal + CIoU + WMMA block reduction.
// ---------------------------------------------------------------------------
__global__ void loss_o2m_kernel(const float* __restrict__ logits,
                                const float* __restrict__ boxes,
                                const float* __restrict__ gt,
                                const int*   __restrict__ mask,
                                const float* __restrict__ thr,
                                float*       __restrict__ partials) {
  __shared__ float sx1[Mm], sy1[Mm], sx2[Mm], sy2[Mm], sthr[Mm];
  __shared__ float sred[NWAVES * 3];
  int blk = blockIdx.x;
  int b = blk / CHUNKS, chunk = blk % CHUNKS;
  int tid = threadIdx.x;
  if (tid < Mm) {
    int gi = b * Mm + tid;
    sx1[tid] = gt[gi * 4 + 0]; sy1[tid] = gt[gi * 4 + 1];
    sx2[tid] = gt[gi * 4 + 2]; sy2[tid] = gt[gi * 4 + 3];
    sthr[tid] = (mask[gi] > 0) ? fmaxf(thr[gi], 0.f) : 3.0e38f;  // invalid: never cand
  }
  __syncthreads();

  int n = chunk * THREADS + tid;
  float wvalid = (n < Nn) ? 1.f : 0.f;
  int nn = min(n, Nn - 1);
  int idx = b * Nn + nn;
  float lg = logits[idx];
  float4 pb = ((const float4*)boxes)[idx];
  float sc = fminf(fmaxf(1.f / (1.f + __expf(-lg)), EPS_AS_F), 1.f);

  float best = 0.f, biou = 0.f; int bm = 0;
  #pragma unroll 8
  for (int m = 0; m < Mm; ++m) {
    float iou = pair_iou(pb.x, pb.y, pb.z, pb.w, sx1[m], sy1[m], sx2[m], sy2[m]);
    float metric = metric_of(sc, iou);
    bool cand = (iou > EPS_AS_F) && (metric >= sthr[m]) && (metric > 0.f);
    float score = cand ? metric : 0.f;
    if (score > best) { best = score; bm = m; biou = iou; }  // first-occurrence argmax
  }
  float fg = (best > EPS_AS_F) ? wvalid : 0.f;
  float target = fg * biou;                                   // assigned score
  float4 ab; ab.x = fg * sx1[bm]; ab.y = fg * sy1[bm];
             ab.z = fg * sx2[bm]; ab.w = fg * sy2[bm];

  float cls = vfl(lg, target) * wvalid;
  float bx  = ciou_loss_f(pb, ab) * fg;

  float scls, sbox, snfg;
  block_sum3(cls, bx, fg, sred, scls, sbox, snfg);
  if (tid == 0) {
    partials[blk * 3 + 0] = scls;
    partials[blk * 3 + 1] = sbox;
    partials[blk * 3 + 2] = snfg;
  }
}

// ---------------------------------------------------------------------------
// o2o head stage 1: assignment + per-GT best-pred via packed u64 atomicMax
// (value bits << 32 | (0xFFFFFFFF - n)  -> ties pick lowest n, like argmax).
// ---------------------------------------------------------------------------
__global__ void assign_o2o_kernel(const float* __restrict__ logits,
                                  const float* __restrict__ boxes,
                                  const float* __restrict__ gt,
                                  const int*   __restrict__ mask,
                                  const float* __restrict__ thr,
                                  float* __restrict__ maxsc,
                                  int*   __restrict__ maxgt,
                                  float* __restrict__ bioua,
                                  unsigned long long* __restrict__ bestGT) {
  __shared__ float sx1[Mm], sy1[Mm], sx2[Mm], sy2[Mm], sthr[Mm];
  int blk = blockIdx.x;
  int b = blk / CHUNKS, chunk = blk % CHUNKS;
  int tid = threadIdx.x;
  if (tid < Mm) {
    int gi = b * Mm + tid;
    sx1[tid] = gt[gi * 4 + 0]; sy1[tid] = gt[gi * 4 + 1];
    sx2[tid] = gt[gi * 4 + 2]; sy2[tid] = gt[gi * 4 + 3];
    sthr[tid] = (mask[gi] > 0) ? fmaxf(thr[gi], 0.f) : 3.0e38f;
  }
  __syncthreads();

  int n = chunk * THREADS + tid;
  int nn = min(n, Nn - 1);
  int idx = b * Nn + nn;
  float lg = logits[idx];
  float4 pb = ((const float4*)boxes)[idx];
  float sc = fminf(fmaxf(1.f / (1.f + __expf(-lg)), EPS_AS_F), 1.f);

  float best = 0.f, biou = 0.f; int bm = 0;
  #pragma unroll 8
  for (int m = 0; m < Mm; ++m) {
    float iou = pair_iou(pb.x, pb.y, pb.z, pb.w, sx1[m], sy1[m], sx2[m], sy2[m]);
    float metric = metric_of(sc, iou);
    bool cand = (iou > EPS_AS_F) && (metric >= sthr[m]) && (metric > 0.f);
    float score = cand ? metric : 0.f;
    if (score > best) { best = score; bm = m; biou = iou; }
  }
  if (n < Nn) {
    maxsc[idx] = best; maxgt[idx] = bm; bioua[idx] = biou;
    if (best > EPS_AS_F) {
      unsigned long long p = (((unsigned long long)__float_as_uint(best)) << 32) |
                             (unsigned long long)(0xFFFFFFFFu - (unsigned)nn);
      atomicMax(&bestGT[b * Mm + bm], p);
    }
  }
}

// ---------------------------------------------------------------------------
// o2o head stage 2: one-to-one filter + losses + WMMA block reduction.
// ---------------------------------------------------------------------------
__global__ void loss_o2o_kernel(const float* __restrict__ logits,
                                const float* __restrict__ boxes,
                                const float* __restrict__ gt,
                                const float* __restrict__ maxsc,
                                const int*   __restrict__ maxgt,
                                const float* __restrict__ bioua,
                                const unsigned long long* __restrict__ bestGT,
                                float* __restrict__ partials) {
  __shared__ float sx1[Mm], sy1[Mm], sx2[Mm], sy2[Mm];
  __shared__ float sred[NWAVES * 3];
  int blk = blockIdx.x;
  int b = blk / CHUNKS, chunk = blk % CHUNKS;
  int tid = threadIdx.x;
  if (tid < Mm) {
    int gi = b * Mm + tid;
    sx1[tid] = gt[gi * 4 + 0]; sy1[tid] = gt[gi * 4 + 1];
    sx2[tid] = gt[gi * 4 + 2]; sy2[tid] = gt[gi * 4 + 3];
  }
  __syncthreads();

  int n = chunk * THREADS + tid;
  float wvalid = (n < Nn) ? 1.f : 0.f;
  int nn = min(n, Nn - 1);
  int idx = b * Nn + nn;
  float best = maxsc[idx];
  int   bm   = maxgt[idx];
  float biou = bioua[idx];

  float fg = (best > EPS_AS_F) ? wvalid : 0.f;
  if (fg > 0.f) {
    unsigned long long w = bestGT[b * Mm + bm];
    if ((unsigned)(w & 0xFFFFFFFFull) != (0xFFFFFFFFu - (unsigned)nn)) fg = 0.f;
  }

  float lg = logits[idx];
  float4 pb = ((const float4*)boxes)[idx];
  float target = fg * biou;
  float4 ab; ab.x = fg * sx1[bm]; ab.y = fg * sy1[bm];
             ab.z = fg * sx2[bm]; ab.w = fg * sy2[bm];

  float cls = vfl(lg, target) * wvalid;
  float bx  = ciou_loss_f(pb, ab) * fg;

  float scls, sbox, snfg;
  block_sum3(cls, bx, fg, sred, scls, sbox, snfg);
  if (tid == 0) {
    partials[blk * 3 + 0] = scls;
    partials[blk * 3 + 1] = sbox;
    partials[blk * 3 + 2] = snfg;
  }
}

__global__ void init_kernel(unsigned long long* __restrict__ bestGT) {
  int i = blockIdx.x * THREADS + threadIdx.x;
  if (i < Bb * Mm) bestGT[i] = 0ull;
}

__global__ void finalize_kernel(const float* __restrict__ p0,
                                const float* __restrict__ p1,
                                float* __restrict__ out) {
  __shared__ float sred[NWAVES * 3];
  float c0 = 0.f, b0 = 0.f, n0 = 0.f, c1 = 0.f, b1 = 0.f, n1 = 0.f;
  for (int i = threadIdx.x; i < NBLK; i += THREADS) {
    c0 += p0[i * 3 + 0]; b0 += p0[i * 3 + 1]; n0 += p0[i * 3 + 2];
    c1 += p1[i * 3 + 0]; b1 += p1[i * 3 + 1]; n1 += p1[i * 3 + 2];
  }
  float sc0, sb0, sn0, sc1, sb1, sn1;
  block_sum3(c0, b0, n0, sred, sc0, sb0, sn0);
  block_sum3(c1, b1, n1, sred, sc1, sb1, sn1);
  if (threadIdx.x == 0) {
    float nf0 = fmaxf(sn0, 1.f), nf1 = fmaxf(sn1, 1.f);
    out[0] = (sc0 + sb0) / nf0 + (sc1 + sb1) / nf1;
  }
}

// ---------------------------------------------------------------------------
extern "C" void kernel_launch(void* const* d_in, const int* in_sizes, int n_in,
                              void* d_out, int out_size, void* d_ws, size_t ws_size,
                              hipStream_t stream) {
  const float* sc_o2m = (const float*)d_in[0];
  const float* bx_o2m = (const float*)d_in[1];
  const float* sc_o2o = (const float*)d_in[2];
  const float* bx_o2o = (const float*)d_in[3];
  const float* gt     = (const float*)d_in[4];
  const int*   mask   = (const int*)d_in[6];
  float* out = (float*)d_out;

  char* ws = (char*)d_ws;
  const size_t BN4 = (size_t)Bb * Nn * 4;                  // 537600 B
  float*              thr0   = (float*)(ws + 0);           // B*M floats
  float*              thr1   = (float*)(ws + 4096);
  unsigned long long* bestGT = (unsigned long long*)(ws + 8192);
  float*              part0  = (float*)(ws + 16384);       // NBLK*3
  float*              part1  = (float*)(ws + 24576);
  float*              maxsc  = (float*)(ws + 32768);
  int*                maxgt  = (int*)  (ws + 32768 + BN4);
  float*              bioua  = (float*)(ws + 32768 + 2 * BN4);

  init_kernel<<<(Bb * Mm + THREADS - 1) / THREADS, THREADS, 0, stream>>>(bestGT);

  topk_thr_kernel<10><<<Bb * Mm, THREADS, 0, stream>>>(sc_o2m, bx_o2m, gt, mask, thr0);
  topk_thr_kernel<1> <<<Bb * Mm, THREADS, 0, stream>>>(sc_o2o, bx_o2o, gt, mask, thr1);

  loss_o2m_kernel<<<NBLK, THREADS, 0, stream>>>(sc_o2m, bx_o2m, gt, mask, thr0, part0);

  assign_o2o_kernel<<<NBLK, THREADS, 0, stream>>>(sc_o2o, bx_o2o, gt, mask, thr1,
                                                  maxsc, maxgt, bioua, bestGT);
  loss_o2o_kernel<<<NBLK, THREADS, 0, stream>>>(sc_o2o, bx_o2o, gt,
                                                maxsc, maxgt, bioua, bestGT, part1);

  finalize_kernel<<<1, THREADS, 0, stream>>>(part0, part1, out);
}